// JointEdgeLoss_72524817760728
// MI455X (gfx1250) — compile-verified
//
#include <hip/hip_runtime.h>
#include <hip/hip_bf16.h>
#include <stdint.h>

// ---------------- problem geometry ----------------
#define IMG_W 4096
#define IMG_H 4096
#define TW 128                 // tile width  (output pixels)
#define TH 16                  // tile height (output pixels)
#define HALO_W 136             // loaded width: baseX = tileX*TW - 4, 16B aligned
#define HALO_H 18              // loaded rows:  baseY = tileY*TH - 1
#define LDSW 140               // LDS row stride in floats (560B, 16B multiple)
#define NTILE_X (IMG_W / TW)   // 32
#define NTILE_Y (IMG_H / TH)   // 256
#define N_EDGE_PART (NTILE_X * NTILE_Y)  // 8192 block partials
#define N_KPT 1024
#define KPT_BLOCKS 4
#define BLOCK 256

// ---------------- CDNA5 feature probes (device pass only) ----------------
#if defined(__AMDGCN__)
  #if __has_builtin(__builtin_amdgcn_global_load_async_to_lds_b128)
    #define HAVE_ASYNC_LDS 1
  #else
    #define HAVE_ASYNC_LDS 0
    #warning "CDNA5 probe: __builtin_amdgcn_global_load_async_to_lds_b128 NOT available"
  #endif
  #if __has_builtin(__builtin_amdgcn_wmma_f32_16x16x4_f32)
    #define HAVE_WMMA_F32X4 1
  #else
    #define HAVE_WMMA_F32X4 0
    #warning "CDNA5 probe: __builtin_amdgcn_wmma_f32_16x16x4_f32 NOT available"
  #endif
  #if __has_builtin(__builtin_amdgcn_s_wait_asynccnt)
    #define HAVE_WAIT_ASYNC_BUILTIN 1
  #else
    #define HAVE_WAIT_ASYNC_BUILTIN 0
  #endif
#else
  #define HAVE_ASYNC_LDS 0
  #define HAVE_WMMA_F32X4 0
  #define HAVE_WAIT_ASYNC_BUILTIN 0
#endif

typedef __attribute__((ext_vector_type(2))) float v2f;
typedef __attribute__((ext_vector_type(8))) float v8f;

#if HAVE_ASYNC_LDS
// Builtin signature (from clang diagnostic): arg0 is
// 'int __attribute__((vector_size(16))) __attribute__((address_space(1)))*'.
typedef int v4i_vs __attribute__((vector_size(4 * sizeof(int))));
typedef __attribute__((address_space(1))) v4i_vs GV4;
typedef __attribute__((address_space(3))) v4i_vs LV4;

__device__ __forceinline__ void async_copy_b128(const float* gsrc, float* ldst) {
  // flat->AS3 is defined as low-32-bit truncation on AMDGPU
  __builtin_amdgcn_global_load_async_to_lds_b128(
      (GV4*)(uintptr_t)gsrc,
      (LV4*)(uint32_t)(uintptr_t)ldst,
      /*offset=*/0, /*cpol=*/0);
}
__device__ __forceinline__ void wait_async_zero() {
#if HAVE_WAIT_ASYNC_BUILTIN
  __builtin_amdgcn_s_wait_asynccnt(0);
#else
  asm volatile("s_wait_asynccnt 0" ::: "memory");
#endif
}
#endif

// ================= Kernel A: fused Sobel-edge MSE (per-tile partial sums) ==========
// Separable Sobel: ex = vsmooth(hdiff), ey = vdiff(hsmooth). Each thread owns an
// 8-tall column strip; LDS is read 3 floats per row per image instead of 9.
__global__ void edge_loss_kernel(const float* __restrict__ predSeg,
                                 const float* __restrict__ tgtSeg,
                                 float* __restrict__ edgePartial) {
  __shared__ __align__(16) float ldsP[HALO_H * LDSW];
  __shared__ __align__(16) float ldsT[HALO_H * LDSW];
  __shared__ float red[BLOCK];

  const int tid = threadIdx.x;
  const int tileX = blockIdx.x, tileY = blockIdx.y;
  const int baseX = tileX * TW - 4;   // 16B-aligned global column
  const int baseY = tileY * TH - 1;

  const bool interior = (baseX >= 0) && (baseX + HALO_W <= IMG_W) &&
                        (baseY >= 0) && (baseY + HALO_H <= IMG_H);
  (void)interior;

#if HAVE_ASYNC_LDS
  if (interior) {
    const int nChunks = HALO_H * (HALO_W / 4);   // 612 x 16B per image
    for (int c = tid; c < nChunks; c += BLOCK) {
      const int row = c / (HALO_W / 4);
      const int col = (c % (HALO_W / 4)) * 4;
      const size_t goff = (size_t)(baseY + row) * IMG_W + (size_t)(baseX + col);
      async_copy_b128(predSeg + goff, &ldsP[row * LDSW + col]);
      async_copy_b128(tgtSeg  + goff, &ldsT[row * LDSW + col]);
    }
    wait_async_zero();
  } else
#endif
  {
    for (int c = tid; c < HALO_H * HALO_W; c += BLOCK) {
      const int row = c / HALO_W, col = c % HALO_W;
      const int gx = baseX + col, gy = baseY + row;
      float vp = 0.f, vt = 0.f;
      if (gx >= 0 && gx < IMG_W && gy >= 0 && gy < IMG_H) {
        const size_t g = (size_t)gy * IMG_W + (size_t)gx;
        vp = predSeg[g];
        vt = tgtSeg[g];
      }
      ldsP[row * LDSW + col] = vp;
      ldsT[row * LDSW + col] = vt;
    }
  }
  __syncthreads();

  const int x  = tid & (TW - 1);      // 0..127
  const int ys = (tid >> 7) * 8;      // 0 or 8 : first LDS row of the strip
  const int lx = x + 4;               // pixel column in LDS

  float dP[10], sP[10], dT[10], sT[10];
#pragma unroll
  for (int r = 0; r < 10; ++r) {
    const float* rowP = &ldsP[(ys + r) * LDSW];
    const float a = rowP[lx - 1], b = rowP[lx], c = rowP[lx + 1];
    dP[r] = c - a;
    sP[r] = (a + c) + 2.f * b;
    const float* rowT = &ldsT[(ys + r) * LDSW];
    const float at = rowT[lx - 1], bt = rowT[lx], ct = rowT[lx + 1];
    dT[r] = ct - at;
    sT[r] = (at + ct) + 2.f * bt;
  }

  float acc = 0.f;
#pragma unroll
  for (int i = 0; i < 8; ++i) {
    const int r = i + 1;
    const float exP = dP[r - 1] + 2.f * dP[r] + dP[r + 1];
    const float eyP = sP[r + 1] - sP[r - 1];
    const float pe  = sqrtf(exP * exP + eyP * eyP);
    const float exT = dT[r - 1] + 2.f * dT[r] + dT[r + 1];
    const float eyT = sT[r + 1] - sT[r - 1];
    const float te  = sqrtf(exT * exT + eyT * eyT);
    const float d   = pe - te;
    acc += d * d;
  }

  red[tid] = acc;
  __syncthreads();
#pragma unroll
  for (int sN = BLOCK / 2; sN > 0; sN >>= 1) {
    if (tid < sN) red[tid] += red[tid + sN];
    __syncthreads();
  }
  if (tid == 0) edgePartial[blockIdx.y * NTILE_X + blockIdx.x] = red[0];
}

// ================= Kernel B: keypoint constraint (no distance_map materialized) ====
// distance_map = box3x3(sobel_mag(pred_seg)); bilinear(align_corners=False, zero pad).
// Each corner needs a 5x5 seg window -> 4 corners share a 6x6 window.
__global__ void kpt_loss_kernel(const float* __restrict__ kp,
                                const float* __restrict__ seg,
                                float* __restrict__ kptPartial) {
  __shared__ float red[BLOCK];
  const int tid = threadIdx.x;
  const int k = blockIdx.x * BLOCK + tid;   // always < 1024

  const float kx = kp[2 * k + 0];
  const float ky = kp[2 * k + 1];
  const float fx = (kx + 1.f) * (IMG_W * 0.5f) - 0.5f;
  const float fy = (ky + 1.f) * (IMG_H * 0.5f) - 0.5f;
  const int x0 = (int)floorf(fx);
  const int y0 = (int)floorf(fy);
  const float wx1 = fx - (float)x0, wx0 = 1.f - wx1;
  const float wy1 = fy - (float)y0, wy0 = 1.f - wy1;

  float s[6][6];
#pragma unroll
  for (int jj = 0; jj < 6; ++jj) {
    const int py = y0 - 2 + jj;
#pragma unroll
    for (int ii = 0; ii < 6; ++ii) {
      const int px = x0 - 2 + ii;
      const bool inb = (px >= 0) && (px < IMG_W) && (py >= 0) && (py < IMG_H);
      float v = 0.f;
      if (inb) v = seg[(size_t)py * IMG_W + (size_t)px];
      s[jj][ii] = v;
    }
  }

  float e[4][4];
#pragma unroll
  for (int j = 0; j < 4; ++j) {
    const int py = y0 - 1 + j;
#pragma unroll
    for (int i = 0; i < 4; ++i) {
      const int px = x0 - 1 + i;
      if (px < 0 || px >= IMG_W || py < 0 || py >= IMG_H) { e[j][i] = 0.f; continue; }
      const float ex = (s[j][i + 2] - s[j][i])
                     + 2.f * (s[j + 1][i + 2] - s[j + 1][i])
                     + (s[j + 2][i + 2] - s[j + 2][i]);
      const float ey = (s[j + 2][i] + 2.f * s[j + 2][i + 1] + s[j + 2][i + 2])
                     - (s[j][i] + 2.f * s[j][i + 1] + s[j][i + 2]);
      e[j][i] = sqrtf(ex * ex + ey * ey);
    }
  }

  float corner[2][2];
#pragma unroll
  for (int b = 0; b < 2; ++b) {
#pragma unroll
    for (int a = 0; a < 2; ++a) {
      const int cx = x0 + a, cy = y0 + b;
      float v = 0.f;
      if (cx >= 0 && cx < IMG_W && cy >= 0 && cy < IMG_H) {
#pragma unroll
        for (int j = b; j < b + 3; ++j)
#pragma unroll
          for (int i = a; i < a + 3; ++i) v += e[j][i];
      }
      corner[b][a] = v;
    }
  }

  const float kd = wx0 * wy0 * corner[0][0] + wx1 * wy0 * corner[0][1]
                 + wx0 * wy1 * corner[1][0] + wx1 * wy1 * corner[1][1];

  red[tid] = kd;
  __syncthreads();
#pragma unroll
  for (int sN = BLOCK / 2; sN > 0; sN >>= 1) {
    if (tid < sN) red[tid] += red[tid + sN];
    __syncthreads();
  }
  if (tid == 0) kptPartial[blockIdx.x] = red[0];
}

// ================= Kernel C: deterministic finalize (WMMA row-sum tail) ============
__global__ void finalize_kernel(const float* __restrict__ edgePartial,
                                const float* __restrict__ kptPartial,
                                float* __restrict__ out) {
  __shared__ float red[BLOCK];
  const int tid = threadIdx.x;

  float a = 0.f;
  for (int i = tid; i < N_EDGE_PART; i += BLOCK) a += edgePartial[i];
  red[tid] = a;
  __syncthreads();
  if (tid < 128) red[tid] += red[tid + 128];
  __syncthreads();
  if (tid < 64) red[tid] += red[tid + 64];
  __syncthreads();
  // red[0..63] hold the remaining partials; reduce them with one
  // V_WMMA_F32_16X16X4_F32: A[m][k] = red[4m+k], B = ones  =>  D rows = row sums.
  float edgeSum = 0.f;
#if HAVE_WMMA_F32X4
  {
    const int lane  = tid & 31;
    const int m     = lane & 15;
    const int kbase = (lane >> 4) * 2;          // lanes 0-15: K=0,1; lanes 16-31: K=2,3
    v2f A; A.x = red[m * 4 + kbase]; A.y = red[m * 4 + kbase + 1];
    v2f B; B.x = 1.0f; B.y = 1.0f;
    v8f C = {};
    v8f D = __builtin_amdgcn_wmma_f32_16x16x4_f32(false, A, false, B, (short)0, C,
                                                  false, false);
    float sLocal = D[0] + D[1] + D[2] + D[3] + D[4] + D[5] + D[6] + D[7];
    // lanes 0-15 hold rows 0-7 (sum of red[0..31]); lanes 16-31 rows 8-15.
    edgeSum = sLocal + __shfl_xor(sLocal, 16, 32);
  }
#else
  if (tid == 0) {
    for (int i = 0; i < 64; ++i) edgeSum += red[i];
  }
#endif

  if (tid == 0) {
    const float ks = kptPartial[0] + kptPartial[1] + kptPartial[2] + kptPartial[3];
    const float edge_loss  = edgeSum / 16777216.0f;        // mean over 4096*4096
    const float constraint = ks / (float)N_KPT;
    out[0] = 1.0f * edge_loss + 0.5f * constraint;         // MASK_W, KPT_W
  }
}

// ================= host-side launch ================================================
extern "C" void kernel_launch(void* const* d_in, const int* in_sizes, int n_in,
                              void* d_out, int out_size, void* d_ws, size_t ws_size,
                              hipStream_t stream) {
  (void)in_sizes; (void)n_in; (void)out_size; (void)ws_size;
  const float* predKp  = (const float*)d_in[0];   // [1024,2]
  // d_in[1] (target_keypoints) is unused by the reference computation
  const float* predSeg = (const float*)d_in[2];   // [1,1,4096,4096]
  const float* tgtSeg  = (const float*)d_in[3];   // [1,1,4096,4096]
  float* out = (float*)d_out;

  float* edgePartial = (float*)d_ws;              // 8192 floats
  float* kptPartial  = edgePartial + N_EDGE_PART; // 4 floats

  edge_loss_kernel<<<dim3(NTILE_X, NTILE_Y), BLOCK, 0, stream>>>(predSeg, tgtSeg,
                                                                 edgePartial);
  kpt_loss_kernel<<<KPT_BLOCKS, BLOCK, 0, stream>>>(predKp, predSeg, kptPartial);
  finalize_kernel<<<1, BLOCK, 0, stream>>>(edgePartial, kptPartial, out);
}